// PerClassECE_18141941858508
// MI455X (gfx1250) — compile-verified
//
#include <hip/hip_runtime.h>
#include <hip/hip_bf16.h>
#include <stdint.h>

#define NBINS 15
#define NCLS  100
#define ROWS_PER_TILE 32                 // tile = 32 rows x 400 B = 12.8 KB
#define CHUNKS_PER_TILE (ROWS_PER_TILE * 25)   // 800 x 16B chunks
#define THREADS 256
#define STAGE_OPS 4                      // ceil(800/256): async b128 ops per wave per tile

// ---------------------------------------------------------------------------
// Async global -> LDS staging (CDNA5 GLOBAL_LOAD_ASYNC_TO_LDS_B128, ASYNCcnt).
// Every wave issues exactly STAGE_OPS instructions (tail chunks are clamped to
// the last valid chunk: duplicate writes of identical data, benign), so a
// constant `s_wait_asynccnt STAGE_OPS` after issuing the *next* tile exactly
// drains the *current* tile (per-wave async loads complete in order).
// ---------------------------------------------------------------------------
__device__ __forceinline__ void stage_tile_async(const float* gbase,
                                                 uint32_t lds_off,
                                                 int valid_chunks, int tid) {
#pragma unroll
  for (int k = 0; k < STAGE_OPS; ++k) {
    int chunk = tid + k * THREADS;
    if (chunk >= valid_chunks) chunk = valid_chunks - 1;   // per-lane clamp
    uint64_t g = (uint64_t)(uintptr_t)gbase + (uint64_t)chunk * 16u;
    uint32_t l = lds_off + (uint32_t)chunk * 16u;          // LDS byte offset
    asm volatile("global_load_async_to_lds_b128 %0, %1, off"
                 :: "v"(l), "v"(g) : "memory");
  }
}

__global__ void __launch_bounds__(THREADS)
ece_hist_kernel(const float* __restrict__ logits,
                const int* __restrict__ labels,
                float* __restrict__ ws,
                int nrows, int ntiles) {
  __shared__ __align__(16) float tiles[2][ROWS_PER_TILE * NCLS];   // 25.6 KB
  __shared__ float h_cnt[NCLS * NBINS];                            // 18 KB total
  __shared__ float h_conf[NCLS * NBINS];
  __shared__ float h_acc[NCLS * NBINS];

  const int tid = threadIdx.x;
  for (int i = tid; i < NCLS * NBINS; i += THREADS) {
    h_cnt[i] = 0.f; h_conf[i] = 0.f; h_acc[i] = 0.f;
  }

  // LDS byte offsets (flat shared pointer: low 32 bits == LDS offset)
  const uint32_t lds0 = (uint32_t)(uintptr_t)(&tiles[0][0]);
  const uint32_t lds1 = (uint32_t)(uintptr_t)(&tiles[1][0]);

  const int lane = tid & 31;            // wave32
  const int wave = tid >> 5;
  const int q    = lane & 7;            // 8 lanes cooperate on one row
  const int rloc = (wave << 2) + (lane >> 3);   // row 0..31 within tile

  int t = blockIdx.x;
  if (t < ntiles) {
    { // prologue: stage first tile into buffer 0
      int r0 = t * ROWS_PER_TILE;
      int vr = min(ROWS_PER_TILE, nrows - r0);
      stage_tile_async(logits + (size_t)r0 * NCLS, lds0, vr * 25, tid);
    }
    int cb = 0;
    for (; t < ntiles; t += gridDim.x) {
      int tn = t + (int)gridDim.x;
      if (tn < ntiles) {                // kick next tile into the other buffer
        int r0n = tn * ROWS_PER_TILE;
        int vrn = min(ROWS_PER_TILE, nrows - r0n);
        stage_tile_async(logits + (size_t)r0n * NCLS, cb ? lds0 : lds1,
                         vrn * 25, tid);
        asm volatile("s_wait_asynccnt 0x4" ::: "memory"); // current tile done
      } else {
        asm volatile("s_wait_asynccnt 0x0" ::: "memory");
      }
      __syncthreads();                  // all waves' chunks visible

      const float* row = &tiles[cb][rloc * NCLS];
      const int rg = t * ROWS_PER_TILE + rloc;
      const bool valid = rg < nrows;
      float conf = 0.f, accv = 0.f;
      int pred = 0, bin = 0;
      if (valid) {
        // pass 1: max + argmax (first-index tie-break)
        float m = -3.402823466e38f; int mi = 0;
#pragma unroll
        for (int k = 0; k < 4; ++k) {
          int c = q + (k << 3);
          if (c < 25) {
            float4 v = *reinterpret_cast<const float4*>(row + c * 4);
            int c0 = c * 4;
            if (v.x > m) { m = v.x; mi = c0;     }
            if (v.y > m) { m = v.y; mi = c0 + 1; }
            if (v.z > m) { m = v.z; mi = c0 + 2; }
            if (v.w > m) { m = v.w; mi = c0 + 3; }
          }
        }
#pragma unroll
        for (int off = 1; off <= 4; off <<= 1) {   // reduce 8-lane group
          float om = __shfl_xor(m, off, 32);
          int   oi = __shfl_xor(mi, off, 32);
          if (om > m || (om == m && oi < mi)) { m = om; mi = oi; }
        }
        // pass 2: sum of exp(x - m)
        float s = 0.f;
#pragma unroll
        for (int k = 0; k < 4; ++k) {
          int c = q + (k << 3);
          if (c < 25) {
            float4 v = *reinterpret_cast<const float4*>(row + c * 4);
            s += __expf(v.x - m) + __expf(v.y - m) +
                 __expf(v.z - m) + __expf(v.w - m);
          }
        }
#pragma unroll
        for (int off = 1; off <= 4; off <<= 1) s += __shfl_xor(s, off, 32);

        conf = 1.f / s;                 // == max(softmax)
        pred = mi;
        bin  = (int)ceilf(conf * (float)NBINS) - 1;   // searchsorted(left)-1
        bin  = bin < 0 ? 0 : (bin > NBINS - 1 ? NBINS - 1 : bin);
        accv = (labels[rg] == pred) ? 1.f : 0.f;
      }
      if (valid && q == 0) {            // leader lane: LDS float atomics
        int idx = pred * NBINS + bin;
        atomicAdd(&h_cnt[idx],  1.f);
        atomicAdd(&h_conf[idx], conf);
        atomicAdd(&h_acc[idx],  accv);
      }
      __syncthreads();                  // protect buffer reuse next iteration
      cb ^= 1;
    }
  }
  __syncthreads();
  // drain per-block histogram into global workspace
  for (int i = tid; i < NCLS * NBINS; i += THREADS) {
    atomicAdd(&ws[i],                   h_cnt[i]);
    atomicAdd(&ws[NCLS * NBINS + i],    h_conf[i]);
    atomicAdd(&ws[2 * NCLS * NBINS + i], h_acc[i]);
  }
}

__global__ void ece_zero(float* __restrict__ ws, int n) {
  int i = blockIdx.x * blockDim.x + threadIdx.x;
  if (i < n) ws[i] = 0.f;
}

__global__ void ece_finalize(const float* __restrict__ ws,
                             float* __restrict__ out) {
  int c = blockIdx.x * blockDim.x + threadIdx.x;
  if (c < NCLS) {
    const float* cnt = ws;
    const float* cs  = ws + NCLS * NBINS;
    const float* as  = ws + 2 * NCLS * NBINS;
    float tot = 0.f;
    for (int b = 0; b < NBINS; ++b) tot += cnt[c * NBINS + b];
    float denom = fmaxf(tot, 1.f);
    float e = 0.f;
    for (int b = 0; b < NBINS; ++b) {
      float n  = cnt[c * NBINS + b];
      float sc = fmaxf(n, 1.f);
      float gap  = fabsf(cs[c * NBINS + b] / sc - as[c * NBINS + b] / sc);
      float prop = n / denom;
      e += (n > 0.f) ? gap * prop : 0.f;
    }
    out[c] = e;
  }
}

extern "C" void kernel_launch(void* const* d_in, const int* in_sizes, int n_in,
                              void* d_out, int out_size, void* d_ws, size_t ws_size,
                              hipStream_t stream) {
  const float* logits = (const float*)d_in[0];
  const int*   labels = (const int*)d_in[1];
  float* ws  = (float*)d_ws;
  float* out = (float*)d_out;

  const int nrows  = in_sizes[1];                       // N (labels count)
  const int ntiles = (nrows + ROWS_PER_TILE - 1) / ROWS_PER_TILE;

  const int nhist = 3 * NCLS * NBINS;                   // 4500 floats
  ece_zero<<<(nhist + 255) / 256, 256, 0, stream>>>(ws, nhist);

  int grid = ntiles < 1024 ? ntiles : 1024;
  ece_hist_kernel<<<grid, THREADS, 0, stream>>>(logits, labels, ws, nrows, ntiles);

  ece_finalize<<<1, 128, 0, stream>>>(ws, out);
}